// ROIpooling_38474317037641
// MI455X (gfx1250) — compile-verified
//
#include <hip/hip_runtime.h>
#include <cstdint>
#include <cstddef>

#define POOL   7
#define MAXB   4
#define HH     50
#define WW     50
// ROI span after /16 quantization is <= 17 cells (boxes <= 256px), so an
// 18-row x 32-col window clamped inside the 50x50 plane always covers it.
#define TROWS  18
#define TCOLS  32
#define TSTR   33                    // padded LDS row stride (floats)
#define TILEF  (TROWS * TSTR)        // 594 floats per tile
#define WAVES  8                     // 256 threads / wave32
#define NBUF   2                     // double buffer per wave

__global__ __launch_bounds__(256) void roi_pool_cdna5_kernel(
    const float* __restrict__ feat,   // [C, 50, 50]
    const float* __restrict__ rois,   // [N, 4] (x1,y1,x2,y2)
    float* __restrict__ out,          // [N, C, 7, 7]
    int C, int N, int cpb)
{
    __shared__ float tile[WAVES * NBUF * TILEF];   // ~38 KB
    __shared__ int hoff[POOL], hcnt[POOL];
    __shared__ int woff[POOL], wcnt[POOL];
    __shared__ int org[2];                         // [ystart, xstart]

    const int n    = blockIdx.x;
    const int c0   = blockIdx.y * cpb;
    const int tid  = threadIdx.x;
    const int wave = tid >> 5;
    const int lane = tid & 31;

    // ---- one-time per-ROI adaptive-pool bin tables (reference semantics) ----
    if (tid < 2 * POOL) {
        const bool isCol = (tid >= POOL);
        const int  bin   = isCol ? tid - POOL : tid;
        const float blo = rois[n * 4 + (isCol ? 0 : 1)];
        const float bhi = rois[n * 4 + (isCol ? 2 : 3)];
        const int lo = (int)(blo * 0.0625f);       // trunc toward zero, non-negative
        const int hi = (int)(bhi * 0.0625f);
        const int L  = hi - lo + 1;
        const int st = (bin * L) / POOL;
        const int en = ((bin + 1) * L + POOL - 1) / POOL;
        int cnt = en - st; if (cnt > MAXB) cnt = MAXB;     // MAX_BIN cap
        const int span = isCol ? TCOLS : TROWS;
        const int size = isCol ? WW : HH;
        int ws = lo; if (ws > size - span) ws = size - span;
        if (ws < 0) ws = 0;
        if (isCol) { woff[bin] = lo + st - ws; wcnt[bin] = cnt; if (bin == 0) org[1] = ws; }
        else       { hoff[bin] = lo + st - ws; hcnt[bin] = cnt; if (bin == 0) org[0] = ws; }
    }
    __syncthreads();                               // only barrier in the kernel

    const int ystart = org[0];
    const int xstart = org[1];
    const uint32_t lds_base = (uint32_t)(uintptr_t)(&tile[0]);

    // ---- per-lane, channel-invariant element offsets (registers) ----
    // lane handles bins p0 = lane and p1 = lane + 32 (if < 49).
    // Out-of-range kh/kw are clamped to row/col 0 of the bin window: duplicate
    // reads inside the valid hc x wc rectangle leave the max unchanged, so the
    // inner loop needs no masking at all.
    int eoff0[MAXB * MAXB];
    int eoff1[MAXB * MAXB];
    #pragma unroll
    for (int j = 0; j < 2; ++j) {
        const int p  = lane + 32 * j;
        const int pc = p < POOL * POOL ? p : POOL * POOL - 1;   // dummy for idle lanes
        const int ph = pc / POOL;
        const int pw = pc - ph * POOL;
        const int base = hoff[ph] * TSTR + woff[pw];
        const int hc = hcnt[ph];
        const int wc = wcnt[pw];
        #pragma unroll
        for (int kh = 0; kh < MAXB; ++kh) {
            const int r = (kh < hc) ? kh * TSTR : 0;
            #pragma unroll
            for (int kw = 0; kw < MAXB; ++kw) {
                const int cidx = (kw < wc) ? kw : 0;
                const int off = base + r + cidx;
                if (j == 0) eoff0[kh * MAXB + kw] = off;
                else        eoff1[kh * MAXB + kw] = off;
            }
        }
    }
    const bool act1 = (lane + 32) < POOL * POOL;

    const int iters = cpb / WAVES;                 // 8 channels per wave

    // ---- stage channel k's window into buffer b (18 coalesced 128B async rows) ----
    auto stage = [&](int k, int b) {
        const int c = c0 + k * WAVES + wave;
        const float* g = feat + (size_t)c * (HH * WW) + ystart * WW + xstart + lane;
        uint32_t l = lds_base + (uint32_t)(((wave * NBUF + b) * TILEF + lane) * 4);
        #pragma unroll
        for (int i = 0; i < TROWS; ++i) {
            asm volatile("global_load_async_to_lds_b32 %0, %1, off"
                         :: "v"(l), "v"(g) : "memory");
            g += WW;
            l += TSTR * 4;
        }
    };

    int b = 0;
    stage(0, b);
    for (int k = 0; k < iters; ++k) {
        if (k + 1 < iters) {
            stage(k + 1, b ^ 1);                   // prefetch next channel into other buffer
            // async-to-LDS completes in order: <=18 outstanding => buffer b resident
            asm volatile("s_wait_asynccnt 0x12" ::: "memory");
        } else {
            asm volatile("s_wait_asynccnt 0x0" ::: "memory");
        }

        // ---- 16 unconditional clause-able LDS loads + max tree per output ----
        const int c = c0 + k * WAVES + wave;
        const float* tb = &tile[(wave * NBUF + b) * TILEF];
        float v0[MAXB * MAXB], v1[MAXB * MAXB];
        #pragma unroll
        for (int e = 0; e < MAXB * MAXB; ++e) {
            v0[e] = tb[eoff0[e]];
            v1[e] = tb[eoff1[e]];
        }
        #pragma unroll
        for (int s = MAXB * MAXB / 2; s > 0; s >>= 1) {
            #pragma unroll
            for (int e = 0; e < s; ++e) {
                v0[e] = fmaxf(v0[e], v0[e + s]);
                v1[e] = fmaxf(v1[e], v1[e + s]);
            }
        }
        const size_t obase = ((size_t)n * C + c) * (POOL * POOL);
        out[obase + lane] = v0[0];
        if (act1) out[obase + lane + 32] = v1[0];
        b ^= 1;
    }
}

extern "C" void kernel_launch(void* const* d_in, const int* in_sizes, int n_in,
                              void* d_out, int out_size, void* d_ws, size_t ws_size,
                              hipStream_t stream) {
    const float* feat = (const float*)d_in[0];   // (1, C, 50, 50) f32
    const float* rois = (const float*)d_in[1];   // (N, 4) f32
    float* out        = (float*)d_out;           // (N, C, 7, 7) f32

    const int C = in_sizes[0] / (HH * WW);       // 512
    const int N = in_sizes[1] / 4;               // 256

    const int cpb = 64;                          // channels per block (8 per wave)
    dim3 grid(N, (C + cpb - 1) / cpb);           // 256 x 8 blocks
    dim3 block(256);                             // 8 wave32s
    roi_pool_cdna5_kernel<<<grid, block, 0, stream>>>(feat, rois, out, C, N, cpb);
}